// OptimizedRNN_58995670778381
// MI455X (gfx1250) — compile-verified
//
#include <hip/hip_runtime.h>

// CDNA5 / gfx1250, wave32. f32 WMMA 16x16x4:
//   A: 16x4  f32, 2 VGPRs: vgpr0 = K=0 (lanes 0-15) / K=2 (lanes 16-31), vgpr1 = K=1 / K=3
//   B: 4x16  f32, 2 VGPRs: mirrored (row K striped across lanes)
//   C/D: 16x16 f32, 8 VGPRs: vgpr r, lane l -> (M = r + 8*(l>>4), N = l&15)
//
// L2 strategy (192 MB global L2): the 64 MB xW workspace is produced by the
// GEMM kernel and consumed once by the scan -> keep it regular-temporal so it
// stays L2-resident. All pure-streaming traffic (X reads, input-copy writes,
// hidden/output writes) is marked non-temporal so it does not evict xW.

typedef __attribute__((ext_vector_type(2))) float v2f;
typedef __attribute__((ext_vector_type(4))) float v4f;
typedef __attribute__((ext_vector_type(8))) float v8f;

#define BB 256
#define SS 4096
#define DD 64
#define HH 16

__device__ __forceinline__ float swapx16(float x) {
  // ds_swizzle_b32 SWAPX16 (group-of-32: xor_mask=0x10, and_mask=0x1f): lane <-> lane^16
  return __int_as_float(__builtin_amdgcn_ds_swizzle(__float_as_int(x), 0x401f));
}

// ---------------------------------------------------------------------------
// Kernel A: xW[s][b][h] = X[b][s][:] @ W_ih[:,h] + bias[h]   (WMMA f32 16x16x4)
// Fused: streams X through once (NT) and writes the verbatim input copy (NT).
// One wave per (s, 16-batch tile); orientation M=h, N=b so xW stores are 2x b128.
// ---------------------------------------------------------------------------
__global__ void rnn_xw_gemm(const float* __restrict__ X,    // [B,S,D]
                            const float* __restrict__ Wih,  // [D,H]
                            const float* __restrict__ bias, // [H]
                            float* __restrict__ xw,         // [S,B,H] (workspace, L2-resident)
                            float* __restrict__ xcpy)       // [B,S,D] (output #3)
{
  const int lane = threadIdx.x & 31;
  const int wave = threadIdx.x >> 5;
  const int t    = blockIdx.x * 8 + wave;     // 65536 waves total
  const int s    = t >> 4;
  const int bt   = (t & 15) * 16;
  const int half = lane >> 4;
  const int l16  = lane & 15;

  // A chunks = W_ih^T: A_k(m=h=l16, K=4k+j+2*half) = Wih[K][h]  (tiny, cached)
  v2f a[16];
#pragma unroll
  for (int k = 0; k < 16; ++k) {
    const int d0 = 4 * k + 2 * half;
    a[k].x = Wih[(d0 + 0) * HH + l16];
    a[k].y = Wih[(d0 + 1) * HH + l16];
  }

  // C = bias broadcast: c[r] = bias[r + 8*half]
  v8f c;
#pragma unroll
  for (int r = 0; r < 8; ++r) c[r] = bias[r + 8 * half];

  // B chunks = X tile: B_k(K=d=4k+j+2*half, n=b=bt+l16) = X[b][s][d]  (NT stream)
  const float* xrow = X + ((size_t)(bt + l16) * SS + s) * DD;
  float*       crow = xcpy + ((size_t)(bt + l16) * SS + s) * DD;
  v2f b[16];
#pragma unroll
  for (int k = 0; k < 16; ++k) {
    const int d0 = 4 * k + 2 * half;
    b[k] = __builtin_nontemporal_load((const v2f*)(xrow + d0));
  }
  // verbatim input copy (each element touched exactly once across the wave)
#pragma unroll
  for (int k = 0; k < 16; ++k) {
    const int d0 = 4 * k + 2 * half;
    __builtin_nontemporal_store(b[k], (v2f*)(crow + d0));
  }

#pragma unroll
  for (int k = 0; k < 16; ++k)
    c = __builtin_amdgcn_wmma_f32_16x16x4_f32(false, a[k], false, b[k],
                                              (short)0, c, false, false);

  // D(m=h, n=b): lane holds b = bt+l16, h = r+8*half -> xw[s][b][h], 2x b128
  // Regular-temporal: xW should remain resident in the 192 MB L2 for the scan.
  float* orow = xw + ((size_t)s * BB + bt + l16) * HH + 8 * half;
  v4f lo = {c[0], c[1], c[2], c[3]};
  v4f hi = {c[4], c[5], c[6], c[7]};
  *(v4f*)(orow)     = lo;
  *(v4f*)(orow + 4) = hi;
}

// ---------------------------------------------------------------------------
// Kernel B: sequential scan, transposed recurrence G = h^T (16h x 16b):
//   G_new = relu(W_hh^T @ G + xW^T)
// D output is already in B-operand orientation for the next step; the layout
// fixup is 8x ds_swizzle(SWAPX16) + 8x cndmask. The 4 K-chunk WMMAs are
// independent (tree-summed in VALU) to keep the serial chain ~1 WMMA deep.
// One wave per 16-batch tile -> 16 waves, the maximal scan parallelism.
// ---------------------------------------------------------------------------
__global__ void rnn_scan(const float* __restrict__ xw,   // [S,B,H] (L2-resident)
                         const float* __restrict__ Whh,  // [H,H]
                         const float* __restrict__ lw,   // [H]
                         const float* __restrict__ lb,   // [1]
                         float* __restrict__ out,        // [B,S]
                         float* __restrict__ hid)        // [B,S,H]
{
  const int lane = threadIdx.x & 31;
  const int half = lane >> 4;
  const int l16  = lane & 15;
  const int bt   = blockIdx.x * 16;
  const int b    = bt + l16;

  // A chunks = W_hh^T (constant across the scan):
  // A_k(m=h_out=l16, K=h_in=4k+j+2*half) = Whh[h_in][h_out]
  v2f a[4];
#pragma unroll
  for (int k = 0; k < 4; ++k) {
    const int k0 = 4 * k + 2 * half;
    a[k].x = Whh[(k0 + 0) * HH + l16];
    a[k].y = Whh[(k0 + 1) * HH + l16];
  }
  float lwv[8];
#pragma unroll
  for (int r = 0; r < 8; ++r) lwv[r] = lw[r + 8 * half];
  const float lbv = lb[0];

  // G state as B chunks; h0 = 0
  v2f g[4];
#pragma unroll
  for (int k = 0; k < 4; ++k) { g[k].x = 0.f; g[k].y = 0.f; }

  const v8f zc = {};

  for (int s = 0; s < SS; ++s) {
    // C tile: c[r] = xw[s][b][r + 8*half]  (2x b128, contiguous 1KB per wave)
    const float* crow = xw + ((size_t)s * BB + b) * HH + 8 * half;
    __builtin_prefetch(crow + 8 * BB * HH, 0, 0);   // ~8 steps ahead
    v4f clo = *(const v4f*)crow;
    v4f chi = *(const v4f*)(crow + 4);
    v8f c;
#pragma unroll
    for (int r = 0; r < 4; ++r) { c[r] = clo[r]; c[r + 4] = chi[r]; }

    // 4 independent WMMAs (only p0 carries C) -> tree sum + relu
    v8f p0 = __builtin_amdgcn_wmma_f32_16x16x4_f32(false, a[0], false, g[0], (short)0, c,  false, false);
    v8f p1 = __builtin_amdgcn_wmma_f32_16x16x4_f32(false, a[1], false, g[1], (short)0, zc, false, false);
    v8f p2 = __builtin_amdgcn_wmma_f32_16x16x4_f32(false, a[2], false, g[2], (short)0, zc, false, false);
    v8f p3 = __builtin_amdgcn_wmma_f32_16x16x4_f32(false, a[3], false, g[3], (short)0, zc, false, false);

    v8f d;
#pragma unroll
    for (int r = 0; r < 8; ++r)
      d[r] = fmaxf((p0[r] + p1[r]) + (p2[r] + p3[r]), 0.f);

    // hidden[b][s][h]: lane holds b fixed, h = r+8*half -> 2x b128 (NT stream)
    float* hrow = hid + ((size_t)b * SS + s) * HH + 8 * half;
    v4f hlo = {d[0], d[1], d[2], d[3]};
    v4f hhi = {d[4], d[5], d[6], d[7]};
    __builtin_nontemporal_store(hlo, (v4f*)(hrow));
    __builtin_nontemporal_store(hhi, (v4f*)(hrow + 4));

    // output[b][s] = hidden . lw + lb (half-wave reduce via SWAPX16)
    float part = 0.f;
#pragma unroll
    for (int r = 0; r < 8; ++r) part = fmaf(d[r], lwv[r], part);
    const float tot = part + swapx16(part) + lbv;
    if (half == 0) __builtin_nontemporal_store(tot, out + (size_t)b * SS + s);

    // D -> B-operand fixup for next step: per D vgpr, one half stays in place,
    // the other half swaps wave halves (SWAPX16) into a different chunk reg.
    float sw0 = swapx16(d[0]), sw1 = swapx16(d[1]);
    float sw2 = swapx16(d[2]), sw3 = swapx16(d[3]);
    float sw4 = swapx16(d[4]), sw5 = swapx16(d[5]);
    float sw6 = swapx16(d[6]), sw7 = swapx16(d[7]);
    g[0].x = half ? sw2 : d[0];   // K0 / K2
    g[0].y = half ? sw3 : d[1];   // K1 / K3
    g[1].x = half ? sw6 : d[4];   // K4 / K6
    g[1].y = half ? sw7 : d[5];   // K5 / K7
    g[2].x = half ? d[2] : sw0;   // K8 / K10
    g[2].y = half ? d[3] : sw1;   // K9 / K11
    g[3].x = half ? d[6] : sw4;   // K12/ K14
    g[3].y = half ? d[7] : sw5;   // K13/ K15
  }
}

// ---------------------------------------------------------------------------
extern "C" void kernel_launch(void* const* d_in, const int* in_sizes, int n_in,
                              void* d_out, int out_size, void* d_ws, size_t ws_size,
                              hipStream_t stream) {
  const float* X    = (const float*)d_in[0];  // [B,S,D]
  const float* Wih  = (const float*)d_in[1];  // [D,H]
  const float* Whh  = (const float*)d_in[2];  // [H,H]
  const float* bias = (const float*)d_in[3];  // [H]
  const float* lw   = (const float*)d_in[4];  // [1,H]
  const float* lb   = (const float*)d_in[5];  // [1]

  float* out  = (float*)d_out;                         // [B,S,1]
  float* hid  = out + (size_t)BB * SS;                 // [B,S,H]
  float* xcpy = hid + (size_t)BB * SS * HH;            // [B,S,D] (inputs copy)
  float* xw   = (float*)d_ws;                          // [S,B,H] workspace (64 MB)

  // Phase 1: input GEMM + fused input copy. 65536 tiles, 8 waves/block.
  rnn_xw_gemm<<<dim3((SS * (BB / 16)) / 8), dim3(256), 0, stream>>>(X, Wih, bias, xw, xcpy);

  // Phase 2: serial scan, one wave per 16-batch tile.
  rnn_scan<<<dim3(BB / 16), dim3(32), 0, stream>>>(xw, Whh, lw, lb, out, hid);
}